// MLP_46901042872350
// MI455X (gfx1250) — compile-verified
//
#include <hip/hip_runtime.h>

typedef __attribute__((ext_vector_type(2))) float v2f;
typedef __attribute__((ext_vector_type(8))) float v8f;

#define KEXP 21
#define HID  31
#define DIN  32
#define WAVES 4
#define TPW_TARGET 4          // tiles per wave (amortizes W1[k] register load)
#define DESIRED_CAP 16384

// ---------------------------------------------------------------------------
// Kernel 0: zero the per-expert counters (graph-capture-safe, deterministic)
// ---------------------------------------------------------------------------
__global__ void MLP_zero_cnt_kernel(int* __restrict__ cnt) {
    if (threadIdx.x < KEXP) cnt[threadIdx.x] = 0;
}

// ---------------------------------------------------------------------------
// Kernel 1: bucketize rows by t = x[:,0]; zero output; build per-expert lists
// ---------------------------------------------------------------------------
__global__ void MLP_bucket_kernel(const float* __restrict__ x,
                                  int* __restrict__ cnt,
                                  int* __restrict__ lists,
                                  float* __restrict__ out,
                                  int N, int cap) {
    int n = blockIdx.x * blockDim.x + threadIdx.x;
    if (n >= N) return;
    out[n] = 0.0f;                      // rows in no bucket stay zero
    float t = x[(size_t)n * DIN];
    int k;
    if (t > 2.0f) {
        k = 20;                         // edges[20]=2.0, edges[21]=1000
    } else {
        k = (int)ceilf(t * 10.0f) - 1;
        // fix rounding against the exact f32 edges (edge[k] = (float)k / 10)
        if (k >= 0 && k <= 19) {
            float lo = (float)k / 10.0f;
            float hh = (float)(k + 1) / 10.0f;
            if (!(t > lo))       k -= 1;
            else if (!(t <= hh)) k += 1;
        }
    }
    if (k < 0 || k > 20 || !(t > 0.0f)) return;
    int pos = atomicAdd(&cnt[k], 1);
    if (pos < cap) lists[k * cap + pos] = n;
}

// ---------------------------------------------------------------------------
// Kernel 2: per-expert batched GEMM with fp32 WMMA (16x16x4), fused MLP
//   h = leaky(feats @ W1[k] + b1[k]);  out = h . W2[k] + b2[k]
// One wave = 16-row tiles of its expert's row list (grid-stride over tiles).
// W1[k]/b1/W2 live in registers across the whole tile loop.
// ---------------------------------------------------------------------------
__global__ void __launch_bounds__(32 * WAVES)
MLP_expert_kernel(const float* __restrict__ x,
                  const float* __restrict__ a,
                  const float* __restrict__ b,
                  const float* __restrict__ W1,
                  const float* __restrict__ b1,
                  const float* __restrict__ W2,
                  const float* __restrict__ b2,
                  const int*   __restrict__ cnt,
                  const int*   __restrict__ lists,
                  float* __restrict__ out,
                  int cap) {
    const int k    = blockIdx.y;
    const int wave = threadIdx.x >> 5;
    const int lane = threadIdx.x & 31;

    int count = cnt[k];
    if (count > cap) count = cap;

    const int hi   = lane >> 4;   // lane half: K-pair select / output row group
    const int m    = lane & 15;   // A-matrix row within tile / column index
    const int ecol = lane & 15;   // output column within 16-wide tile

    // ---- B matrix (W1[k]: K=d padded to 32, N=e in two 16-wide tiles) -> regs.
    // wmma_f32_16x16x4 B layout: VGPR0 = B[4c+2*hi][e], VGPR1 = B[4c+2*hi+1][e]
    const float* W1k = W1 + (size_t)k * (HID * HID);
    v2f Bm[2][8];
    float biasc[2], w2c[2];
    #pragma unroll
    for (int tc = 0; tc < 2; ++tc) {
        int  e   = tc * 16 + ecol;
        int  ec  = (e < HID) ? e : (HID - 1);     // clamped (branchless)
        bool eok = (e < HID);
        #pragma unroll
        for (int c = 0; c < 8; ++c) {
            int d0  = c * 4 + 2 * hi;             // <= 30 always: in-bounds
            int d1  = d0 + 1;                     // == 31 only for c==7,hi==1
            int d1c = (d1 < HID) ? d1 : (HID - 1);
            float v0 = W1k[(size_t)d0  * HID + ec];
            float v1 = W1k[(size_t)d1c * HID + ec];
            v0 = eok ? v0 : 0.0f;
            v1 = (eok && d1 < HID) ? v1 : 0.0f;
            Bm[tc][c] = (v2f){v0, v1};
        }
        biasc[tc] = eok ? b1[k * HID + ec] : 0.0f;
        w2c[tc]   = eok ? W2[k * HID + ec] : 0.0f;
    }
    const float bias2 = b2[k];

    const int tile0   = blockIdx.x * WAVES + wave;
    const int tstride = gridDim.x * WAVES;

    for (int tile = tile0; tile * 16 < count; tile += tstride) {  // wave-uniform
        const int base = tile * 16;
        int valid = count - base; if (valid > 16) valid = 16;

        // gather row index; clamp OOB lanes to base (valid, never written back)
        int gi = base + m;
        gi = (gi < count) ? gi : base;
        const int rowIdx = lists[k * cap + gi];
        const float* xr = x + (size_t)rowIdx * DIN;

        // ---- hoisted A-side loads + feature FMA (one clause, one wait)
        // lane needs feats[d0], feats[d0+1]; only c==7,hi==1 would touch
        // xr[32]/a[31] (1 word past the buffers) -> shift that window down one
        // and select (f1, 0) branchlessly.
        v2f Apair[8];
        #pragma unroll
        for (int c = 0; c < 8; ++c) {
            int  d0 = c * 4 + 2 * hi;
            bool sh = (c == 7) && (hi == 1);
            int  xo = 1 + d0 - (sh ? 1 : 0);
            int  ao = d0 - (sh ? 1 : 0);
            float p0 = xr[xo],    p1 = xr[xo + 1];
            float a0 = a[ao],     a1 = a[ao + 1];
            float c0 = b[ao],     c1 = b[ao + 1];
            float f0 = p0 * a0 + c0;
            float f1 = p1 * a1 + c1;
            Apair[c].x = sh ? f1 : f0;
            Apair[c].y = sh ? 0.0f : f1;
        }

        // ---- accumulate: C init = b1 broadcast down rows
        v8f acc0, acc1;
        #pragma unroll
        for (int i = 0; i < 8; ++i) { acc0[i] = biasc[0]; acc1[i] = biasc[1]; }

        #pragma unroll
        for (int c = 0; c < 8; ++c) {
            acc0 = __builtin_amdgcn_wmma_f32_16x16x4_f32(false, Apair[c], false, Bm[0][c],
                                                         (short)0, acc0, false, false);
            acc1 = __builtin_amdgcn_wmma_f32_16x16x4_f32(false, Apair[c], false, Bm[1][c],
                                                         (short)0, acc1, false, false);
        }

        // ---- leaky ReLU (slope 0.1 in (0,1) => leaky(h) == max(h, 0.1h),
        // NaN propagates identically) + 2nd-layer dot + 16-lane xor-reduce
        float p[8];
        #pragma unroll
        for (int i = 0; i < 8; ++i) {
            float h0 = fmaxf(acc0[i], 0.1f * acc0[i]);
            float h1 = fmaxf(acc1[i], 0.1f * acc1[i]);
            float v = h0 * w2c[0] + h1 * w2c[1];
            v += __shfl_xor(v, 1, 32);
            v += __shfl_xor(v, 2, 32);
            v += __shfl_xor(v, 4, 32);
            v += __shfl_xor(v, 8, 32);
            p[i] = v;   // row (i + 8*hi), identical on all 16 lanes of the half
        }

        // lane j (j = lane&15 < 8) writes row j + 8*hi; branchless select of p[j]
        float vsel = p[0];
        #pragma unroll
        for (int i = 1; i < 8; ++i) vsel = (m == i) ? p[i] : vsel;
        int mrow = (lane & 7) + 8 * hi;
        if ((m < 8) && (mrow < valid)) {
            int n = lists[k * cap + base + mrow];
            out[n] = vsel + bias2;
        }
    }
}

// ---------------------------------------------------------------------------
extern "C" void kernel_launch(void* const* d_in, const int* in_sizes, int n_in,
                              void* d_out, int out_size, void* d_ws, size_t ws_size,
                              hipStream_t stream) {
    const float* x  = (const float*)d_in[0];
    const float* a  = (const float*)d_in[1];
    const float* b  = (const float*)d_in[2];
    const float* W1 = (const float*)d_in[3];
    const float* b1 = (const float*)d_in[4];
    const float* W2 = (const float*)d_in[5];
    const float* b2 = (const float*)d_in[6];
    float* out = (float*)d_out;

    const int N = in_sizes[0] / DIN;

    // workspace: [256B counters][K * cap row-index lists]
    int* cnt   = (int*)d_ws;
    int* lists = (int*)((char*)d_ws + 256);
    size_t cap_avail = (ws_size > 256) ? (ws_size - 256) / (KEXP * sizeof(int)) : 0;
    int cap = (cap_avail < (size_t)DESIRED_CAP) ? (int)cap_avail : DESIRED_CAP;
    if (cap < 16) cap = 16;  // degenerate-ws fallback; expected path is 16384

    MLP_zero_cnt_kernel<<<1, 32, 0, stream>>>(cnt);
    MLP_bucket_kernel<<<(N + 255) / 256, 256, 0, stream>>>(x, cnt, lists, out, N, cap);

    int tiles   = (cap + 15) / 16;                         // tiles per expert
    int gx      = (tiles + WAVES * TPW_TARGET - 1) / (WAVES * TPW_TARGET);
    if (gx < 1) gx = 1;
    dim3 g2(gx, KEXP);
    MLP_expert_kernel<<<g2, 32 * WAVES, 0, stream>>>(x, a, b, W1, b1, W2, b2,
                                                     cnt, lists, out, cap);
}